// ZCANormOrg_16106127360515
// MI455X (gfx1250) — compile-verified
//
#include <hip/hip_runtime.h>
#include <hip/hip_bf16.h>

typedef __attribute__((ext_vector_type(2))) float v2f;
typedef __attribute__((ext_vector_type(8))) float v8f;

#define HW    3136          // 56*56
#define NCHAN 256
#define NGRP  16
#define MTOT  200704        // 64*3136
#define ZEPS  1e-4f

// ---------------------------------------------------------------------------
// Kernel 0: zero the accumulation workspace (gsum + gram = 4352 floats)
// ---------------------------------------------------------------------------
__global__ void zca_zero(float* __restrict__ p, int n) {
    int i = blockIdx.x * 256 + threadIdx.x;
    if (i < n) p[i] = 0.0f;
}

// ---------------------------------------------------------------------------
// Kernel 1: per-group Gram matrix  G_g = X_g X_g^T  and per-channel sums.
// grid = (49, 16), block = 256 (8 waves). Each block covers 4096 m values;
// each wave covers 512 consecutive m, 4 per WMMA iteration.
// For a Gram matrix the f32 16x16x4 A-layout and B-layout register images
// coincide: lane L holds X[L%16, m + 2*(L/16) + {0,1}] as a float2.
// ---------------------------------------------------------------------------
__global__ void zca_gram(const float* __restrict__ x,
                         float* __restrict__ gram,   // [16][256]
                         float* __restrict__ gsum) { // [256]
    __shared__ float sCov[256];
    __shared__ float sSum[16];
    const int tid = threadIdx.x;
    sCov[tid] = 0.0f;
    if (tid < 16) sSum[tid] = 0.0f;
    __syncthreads();

    const int g    = blockIdx.y;
    const int wave = tid >> 5;
    const int lane = tid & 31;
    const int ch   = lane & 15;   // channel within group
    const int half = lane >> 4;   // 0: K={0,1}, 1: K={2,3}

    int m  = blockIdx.x * 4096 + wave * 512;
    int n  = m / HW;
    int hw = m % HW;              // multiple of 4; tiles never cross n-blocks

    const size_t laneoff = (size_t)ch * HW + (size_t)(2 * half);

    v8f  acc = {};
    float s  = 0.0f;
    for (int it = 0; it < 128; ++it) {
        const float* p = x + ((size_t)n * NCHAN + (size_t)g * 16) * HW
                           + (size_t)hw + laneoff;
        v2f ab = *(const v2f*)p;           // 8B aligned (hw+2*half is even)
        s += ab.x + ab.y;
        // D = A*B + C with A==B  ->  Gram accumulation
        acc = __builtin_amdgcn_wmma_f32_16x16x4_f32(
                  false, ab, false, ab, (short)0, acc, false, false);
        hw += 4;
        if (hw >= HW) { hw = 0; ++n; }
    }

    // reduce within block: C layout -> entry (i = r + 8*half, j = ch)
    atomicAdd(&sSum[ch], s);
#pragma unroll
    for (int r = 0; r < 8; ++r) {
        int i = r + 8 * half;
        atomicAdd(&sCov[i * 16 + ch], acc[r]);
    }
    __syncthreads();

    atomicAdd(&gram[g * 256 + tid], sCov[tid]);
    if (tid < 16) atomicAdd(&gsum[g * 16 + tid], sSum[tid]);
}

// ---------------------------------------------------------------------------
// Kernel 2: per-group cov = G/M - mu mu^T + eps I, then coupled
// Newton-Schulz iteration for cov^{-1/2}. Fold weight into rows and the
// mean subtraction into a per-row constant. grid = 16 blocks, 256 threads.
// ---------------------------------------------------------------------------
__global__ void zca_proj(const float* __restrict__ gram,
                         const float* __restrict__ gsum,
                         const float* __restrict__ weight,
                         const float* __restrict__ bias,
                         float* __restrict__ Aw,      // [16][256]
                         float* __restrict__ cvec) {  // [256]
    const int g   = blockIdx.x;
    const int tid = threadIdx.x;
    const int i   = tid >> 4;
    const int j   = tid & 15;

    __shared__ float mu[16];
    __shared__ float Y[16][17], Z[16][17], T[16][17], Y2[16][17], Z2[16][17];
    __shared__ float red[256];

    if (tid < 16) mu[tid] = gsum[g * 16 + tid] * (1.0f / (float)MTOT);
    __syncthreads();

    float cov = gram[g * 256 + tid] * (1.0f / (float)MTOT)
              - mu[i] * mu[j] + ((i == j) ? ZEPS : 0.0f);

    // Frobenius norm for normalization
    red[tid] = cov * cov;
    __syncthreads();
    for (int s2 = 128; s2 > 0; s2 >>= 1) {
        if (tid < s2) red[tid] += red[tid + s2];
        __syncthreads();
    }
    const float frob = sqrtf(red[0]);

    Y[i][j] = cov / frob;
    Z[i][j] = (i == j) ? 1.0f : 0.0f;
    __syncthreads();

    for (int it = 0; it < 30; ++it) {
        float a = 0.0f;
        for (int k = 0; k < 16; ++k) a += Z[i][k] * Y[k][j];
        T[i][j] = ((i == j) ? 1.5f : 0.0f) - 0.5f * a;
        __syncthreads();
        float ay = 0.0f, az = 0.0f;
        for (int k = 0; k < 16; ++k) {
            ay += Y[i][k] * T[k][j];
            az += T[i][k] * Z[k][j];
        }
        Y2[i][j] = ay; Z2[i][j] = az;
        __syncthreads();
        Y[i][j] = Y2[i][j]; Z[i][j] = Z2[i][j];
        __syncthreads();
    }

    // cov^{-1/2} = Z / sqrt(frob); fold weight into rows
    const float proj = Z[i][j] * rsqrtf(frob);
    const float aw   = weight[g * 16 + i] * proj;
    Aw[g * 256 + tid] = aw;

    red[tid] = aw * mu[j];
    __syncthreads();
    if (j == 0) {
        float d = 0.0f;
        for (int k = 0; k < 16; ++k) d += red[i * 16 + k];
        cvec[g * 16 + i] = bias[g * 16 + i] - d;
    }
}

// ---------------------------------------------------------------------------
// Kernel 3: out[i,m] = sum_k Aw[i,k] * x[k,m] + cvec[i] via 4 chained WMMAs
// per 16(ch) x 16(m) tile. grid = (1568, 16), block = 256 (8 waves, one
// 16-m tile per wave). 16 | 3136 so tiles never cross image boundaries.
// ---------------------------------------------------------------------------
__global__ void zca_apply(const float* __restrict__ x,
                          const float* __restrict__ Aw,
                          const float* __restrict__ cvec,
                          float* __restrict__ out) {
    const int g    = blockIdx.y;
    const int tid  = threadIdx.x;
    const int wave = tid >> 5;
    const int lane = tid & 31;
    const int col  = lane & 15;   // m within tile
    const int half = lane >> 4;

    const int m0  = blockIdx.x * 128 + wave * 16;
    const int n   = m0 / HW;
    const int hw0 = m0 % HW;

    // A tile (weighted projection), A-layout: lane holds row=col, K pair
    const float* AwG = Aw + g * 256;
    v2f a[4];
#pragma unroll
    for (int kb = 0; kb < 4; ++kb)
        a[kb] = *(const v2f*)(AwG + col * 16 + kb * 4 + 2 * half);

    // C init from per-row constant (bias - Aw*mu), C-layout row = r + 8*half
    const float* cv = cvec + g * 16;
    v8f c;
#pragma unroll
    for (int r = 0; r < 8; ++r) c[r] = cv[r + 8 * half];

    const size_t base = ((size_t)n * NCHAN + (size_t)g * 16) * HW
                      + (size_t)hw0 + (size_t)col;

#pragma unroll
    for (int kb = 0; kb < 4; ++kb) {
        const int k0 = kb * 4 + 2 * half;  // B-layout: lane holds K pair, col
        v2f b;
        b.x = x[base + (size_t)k0 * HW];
        b.y = x[base + (size_t)(k0 + 1) * HW];
        c = __builtin_amdgcn_wmma_f32_16x16x4_f32(
                false, a[kb], false, b, (short)0, c, false, false);
    }

#pragma unroll
    for (int r = 0; r < 8; ++r)
        out[base + (size_t)(r + 8 * half) * HW] = c[r];
}

// ---------------------------------------------------------------------------
extern "C" void kernel_launch(void* const* d_in, const int* in_sizes, int n_in,
                              void* d_out, int out_size, void* d_ws, size_t ws_size,
                              hipStream_t stream) {
    const float* x      = (const float*)d_in[0];
    const float* weight = (const float*)d_in[1];
    const float* bias   = (const float*)d_in[2];
    float*       out    = (float*)d_out;

    float* ws   = (float*)d_ws;
    float* gsum = ws;          // 256
    float* gram = ws + 256;    // 4096
    float* Aw   = ws + 4352;   // 4096
    float* cvec = ws + 8448;   // 256

    zca_zero<<<dim3(17), 256, 0, stream>>>(ws, 4352);
    zca_gram<<<dim3(49, 16), 256, 0, stream>>>(x, gram, gsum);
    zca_proj<<<dim3(16), 256, 0, stream>>>(gram, gsum, weight, bias, Aw, cvec);
    zca_apply<<<dim3(1568, 16), 256, 0, stream>>>(x, Aw, cvec, out);
}